// focal_loss_47923245089053
// MI455X (gfx1250) — compile-verified
//
#include <hip/hip_runtime.h>

// Focal loss of the LAST sample only (reference's loop-scoped reset makes all
// other rows dead). Reads one 1000-float row + one label, writes one float.
// Single workgroup, latency-bound: 8 wave32s on one WGP.

typedef __attribute__((ext_vector_type(2))) float v2f;
typedef __attribute__((ext_vector_type(8))) float v8f;
typedef __attribute__((ext_vector_type(4))) int v4i;

#ifndef __has_builtin
#define __has_builtin(x) 0
#endif

#if __has_builtin(__builtin_amdgcn_global_load_async_to_lds_b128)
#define HAVE_ASYNC_LDS 1
// Toolchain signature (from diagnostic): (v4i AS1*, v4i AS3*, Ii offset, Ii cpol)
typedef __attribute__((address_space(1))) v4i gv4i;
typedef __attribute__((address_space(3))) v4i lv4i;
#endif

#define MAX_C 1024  // reference C = 1000; padded to multiple of 64 for WMMA

__global__ __launch_bounds__(256) void focal_last_row_kernel(
    const float* __restrict__ outputs, const int* __restrict__ labels,
    float* __restrict__ out, int B, int C) {
  __shared__ __align__(16) float vals[MAX_C];
  __shared__ float wred[8];
  __shared__ float smax_s;
  __shared__ float ssum_s;

  const int tid = threadIdx.x;
  const int padC = ((C + 63) >> 6) << 6;  // multiple of 64, <= MAX_C here
  const float* row = outputs + (size_t)(B - 1) * (size_t)C;
  const float NEG_INF = -__builtin_inff();

  // ---- 1) Stage last row into LDS (gfx1250 async DMA path when aligned) ----
  bool vec_ok = ((C & 3) == 0) &&
                (((((size_t)(B - 1) * (size_t)C)) & 3) == 0) &&
                ((((size_t)outputs) & 15) == 0);
#ifdef HAVE_ASYNC_LDS
  if (vec_ok) {
    const int nvec = C >> 2;
    for (int v = tid; v < nvec; v += 256) {
      __builtin_amdgcn_global_load_async_to_lds_b128(
          (gv4i*)(row + 4 * v), (lv4i*)(&vals[4 * v]),
          /*offset=*/0, /*cpol=*/0);
    }
    for (int i = C + tid; i < padC; i += 256) vals[i] = NEG_INF;  // pad
#if __has_builtin(__builtin_amdgcn_s_wait_asynccnt)
    __builtin_amdgcn_s_wait_asynccnt(0);
#else
    asm volatile("s_wait_asynccnt 0" ::: "memory");
#endif
  } else
#else
  (void)vec_ok;
#endif
  {
    for (int i = tid; i < padC; i += 256)
      vals[i] = (i < C) ? row[i] : NEG_INF;
  }
  __syncthreads();

  // ---- 2) Row max (wave32 xor-shuffles + 8-wave LDS tree) ----
  float m = NEG_INF;
  for (int i = tid; i < padC; i += 256) m = fmaxf(m, vals[i]);
#pragma unroll
  for (int off = 16; off >= 1; off >>= 1)
    m = fmaxf(m, __shfl_xor(m, off, 32));
  if ((tid & 31) == 0) wred[tid >> 5] = m;
  __syncthreads();
  if (tid == 0) {
    float mm = wred[0];
#pragma unroll
    for (int w = 1; w < 8; ++w) mm = fmaxf(mm, wred[w]);
    smax_s = mm;
  }
  __syncthreads();
  const float mrow = smax_s;

  // ---- 3) exp(x - max) in place (pad: exp(-inf) == 0, inert below) ----
  for (int i = tid; i < padC; i += 256) vals[i] = expf(vals[i] - mrow);
  __syncthreads();

  // ---- 4) Sum of exps: wave 0 via V_WMMA_F32_16X16X4_F32 ----
  // A = ones(16x4) => D[i][n] = sum_k B[k][n]; with C-accumulation over
  // padC/64 issues every element lands in exactly one B cell, so the lane->cell
  // mapping is irrelevant for the total.  D VGPR0: lanes0-15 = (M=0,N=lane),
  // lanes16-31 = (M=8,N=lane-16), identical values; xor-reduce 8..1 totals it.
  if (tid < 32) {  // uniform per-wave branch -> EXEC all ones for WMMA
    float total;
#if __has_builtin(__builtin_amdgcn_wmma_f32_16x16x4_f32)
    v8f acc = {};
    v2f a;
    a[0] = 1.0f;
    a[1] = 1.0f;
    for (int base = 0; base < padC; base += 64) {
      v2f b;
      b[0] = vals[base + tid];
      b[1] = vals[base + 32 + tid];
      acc = __builtin_amdgcn_wmma_f32_16x16x4_f32(
          /*neg_a=*/false, a, /*neg_b=*/false, b,
          /*c_mod=*/(short)0, acc, /*reuse_a=*/false, /*reuse_b=*/false);
    }
    float s = acc[0];
#pragma unroll
    for (int off = 8; off >= 1; off >>= 1) s += __shfl_xor(s, off, 32);
    total = s;
#else
    float s = 0.0f;
    for (int i = tid; i < padC; i += 32) s += vals[i];
#pragma unroll
    for (int off = 16; off >= 1; off >>= 1) s += __shfl_xor(s, off, 32);
    total = s;
#endif
    if (tid == 0) ssum_s = total;
  }
  __syncthreads();

  // ---- 5) Focal loss scalar (row-max cancels in p = e_c / sum) ----
  if (tid == 0) {
    int c = labels[B - 1];
    float p = vals[c] / ssum_s;
    float omp = 1.0f - p;
    float loss;
    if (c == 0)
      loss = -0.1f * omp * omp * logf(p);
    else if (c == 1)
      loss = -0.9f * omp * omp * logf(omp);
    else
      loss = 0.0f;
    out[0] = loss;
  }
}

extern "C" void kernel_launch(void* const* d_in, const int* in_sizes, int n_in,
                              void* d_out, int out_size, void* d_ws,
                              size_t ws_size, hipStream_t stream) {
  const float* outputs = (const float*)d_in[0];
  const int* labels = (const int*)d_in[1];
  const int total = in_sizes[0];   // B * C
  const int B = in_sizes[1];       // 262144
  const int C = total / B;         // 1000 (<= MAX_C)
  focal_last_row_kernel<<<dim3(1), dim3(256), 0, stream>>>(
      outputs, labels, (float*)d_out, B, C);
}